// ReachabilityLoss_12910671692420
// MI455X (gfx1250) — compile-verified
//
#include <hip/hip_runtime.h>
#include <hip/hip_bf16.h>

#define COLS 2048
#define HTOK ((COLS - 2) / 2)   /* 1023 token pairs per row */
#define PAD  1088               /* zero-padded read window [0,1056) + dump [1056,1088) */
#define DUMPC 1056              /* compaction dump slot base */
#define W    4
#define MAX_REACH 5.0f
#define PENALTY_SCALE 1.0f

#define LDS_FENCE() asm volatile("s_wait_dscnt 0x0" ::: "memory")

typedef __attribute__((ext_vector_type(2))) float v2f;
typedef __attribute__((ext_vector_type(8))) float v8f;

__global__ void reach_init_kernel(float* acc) {
    acc[0] = 0.0f;   // sum of per-row penalties
    acc[1] = 0.0f;   // number of valid rows
}

__global__ void reach_fin_kernel(const float* acc, float* out) {
    float c = acc[1];
    out[0] = (c > 0.0f) ? (PENALTY_SCALE * acc[0] / c) : 0.0f;
}

__global__ __launch_bounds__(32)
void reach_row_kernel(const int* __restrict__ ids, float* __restrict__ acc) {
    __shared__ int   raw[COLS];        // async-staged raw row
    __shared__ float xs[PAD];
    __shared__ float ys[PAD];
    __shared__ float ns[PAD];          // |p|^2
    __shared__ float band[16 * W + 32]; // gram band D[m][j-1] + 32 dump slots

    const int lane = threadIdx.x;
    const int row  = blockIdx.x;
    const int* gbase = ids + (size_t)row * COLS;

    // ---- CDNA5 async copy: whole row (8 KB) global -> LDS ----
    unsigned raw_lds =
        (unsigned)(unsigned long long)((__attribute__((address_space(3))) int*)raw);
    #pragma unroll
    for (int it = 0; it < (COLS * 4) / 512; ++it) {
        unsigned off = (unsigned)(it * 512 + lane * 16);
        unsigned dst = raw_lds + off;
        asm volatile("global_load_async_to_lds_b128 %0, %1, %2"
                     :: "v"(dst), "v"(off), "s"(gbase) : "memory");
    }

    // zero-fill compacted arrays while the async copy is in flight
    for (int idx = lane; idx < PAD; idx += 32) {
        xs[idx] = 0.0f; ys[idx] = 0.0f; ns[idx] = 0.0f;
    }

    asm volatile("s_wait_asynccnt 0x0" ::: "memory");

    // ---- wave32 ballot stream compaction of valid holds (branch-free) ----
    int count = 0;
    for (int h0 = 0; h0 < HTOK; h0 += 32) {
        int h  = h0 + lane;
        int hc = (h < HTOK) ? h : (HTOK - 1);      // clamped, always-legal address
        int xt = raw[2 + 2 * hc];
        int yt = raw[3 + 2 * hc];
        bool valid = ((unsigned)(xt - 10) < 11u) &
                     ((unsigned)(yt - 21) < 18u) & (h < HTOK);
        unsigned mask = (unsigned)__ballot(valid);
        int pre = __popc(mask & ((1u << lane) - 1u));
        float xf = (float)(xt - 10);
        float yf = (float)(yt - 20);
        int p = valid ? (count + pre) : (DUMPC + lane);   // select, no branch
        xs[p] = xf; ys[p] = yf; ns[p] = xf * xf + yf * yf;
        count += __popc(mask);
    }
    LDS_FENCE();

    // ---- banded min-distance via WMMA gram tiles ----
    // D1[m][n] = p_{t+m} . p_{t-4+n},  D2 covers the band overhang n=16..18.
    float local = 0.0f;
    const int  l15  = lane & 15;
    const int  half = lane >> 4;
    const bool lo   = lane < 16;
    for (int t = W; t < count; t += 16) {
        // unconditional loads (hi lanes mirror lo lanes), select zeroes K=2,3
        float ax  = xs[t + l15],      ay  = ys[t + l15];
        float b1x = xs[t - 4 + l15],  b1y = ys[t - 4 + l15];
        float b2x = xs[t + 12 + l15], b2y = ys[t + 12 + l15];
        v2f a  = { lo ? ax  : 0.0f, lo ? ay  : 0.0f };
        v2f b1 = { lo ? b1x : 0.0f, lo ? b1y : 0.0f };
        v2f b2 = { lo ? b2x : 0.0f, lo ? b2y : 0.0f };
        v8f c  = {};
        v8f d1 = __builtin_amdgcn_wmma_f32_16x16x4_f32(
                     false, a, false, b1, (short)0, c, false, false);
        v8f d2 = __builtin_amdgcn_wmma_f32_16x16x4_f32(
                     false, a, false, b2, (short)0, c, false, false);

        // branch-free band scatter: invalid pairs go to per-lane dump slots
        #pragma unroll
        for (int r = 0; r < 8; ++r) {
            int m  = r + 8 * half;
            int j1 = m + W - l15;                       // D1: n = l15
            int i1 = ((unsigned)(j1 - 1) < 4u) ? (m * W + j1 - 1) : (64 + lane);
            band[i1] = d1[r];
            int j2 = m + W - 16 - l15;                  // D2: n = 16 + l15
            int i2 = ((unsigned)(j2 - 1) < 4u) ? (m * W + j2 - 1) : (64 + lane);
            band[i2] = d2[r];
        }
        LDS_FENCE();

        // branch-free accumulation (hi lanes compute and are masked out)
        int   i    = t + l15;
        float ni   = ns[i];
        float dmin = 3.4e38f;
        #pragma unroll
        for (int j = 1; j <= W; ++j) {
            float sq = ni + ns[i - j] - 2.0f * band[l15 * W + (j - 1)];
            float dj = (sq > 0.0f) ? sqrtf(sq) : 0.0f;
            dmin = fminf(dmin, dj);
        }
        float viol = fmaxf(dmin - MAX_REACH, 0.0f);
        bool active = lo & (i < count);
        local += active ? viol : 0.0f;
        // WAR to band is safe: LDS pipe is in-order within a wave
    }

    // wave32 reduction
    for (int off = 16; off > 0; off >>= 1)
        local += __shfl_xor(local, off, 32);

    if (lane == 0 && count >= W + 1) {
        atomicAdd(&acc[0], local / (float)(count - W));
        atomicAdd(&acc[1], 1.0f);
    }
}

extern "C" void kernel_launch(void* const* d_in, const int* in_sizes, int n_in,
                              void* d_out, int out_size, void* d_ws, size_t ws_size,
                              hipStream_t stream) {
    const int* ids = (const int*)d_in[0];
    float* acc = (float*)d_ws;
    float* out = (float*)d_out;
    int rows = in_sizes[0] / COLS;

    hipLaunchKernelGGL(reach_init_kernel, dim3(1), dim3(1), 0, stream, acc);
    hipLaunchKernelGGL(reach_row_kernel, dim3(rows), dim3(32), 0, stream, ids, acc);
    hipLaunchKernelGGL(reach_fin_kernel, dim3(1), dim3(1), 0, stream, acc, out);
}